// Point_head_13915694039513
// MI455X (gfx1250) — compile-verified
//
#include <hip/hip_runtime.h>
#include <hip/hip_bf16.h>

typedef __attribute__((ext_vector_type(2))) float v2f;
typedef __attribute__((ext_vector_type(4))) float v4f;
typedef __attribute__((ext_vector_type(8))) float v8f;

#define BVAL 2
#define CCH 96
#define ZD 16
#define YD 256
#define XD 256
#define ZYX (ZD * YD * XD)
#define HCH 32
#define PCH 32
#define DIN 128 /* CCH + PCH */
#define BN_EPS 1e-5f

// ---------------------------------------------------------------------------
// Kernel 1: h = [gather(voxel_feats), point_feat] @ W1 + b1
//   - one wave32 per 16-point tile; D(16x16)=A(16x4,f32)xB(4x16) via
//     v_wmma_f32_16x16x4_f32, 32 K-steps x 2 N-tiles = 64 WMMA per tile
//   - W1 staged in LDS once per block (16 KB)
//   - h written channel-major hT[b][k][n] (128-bit stores)
//   - deterministic per-block partial sum / sumsq per channel for BN stats
// ---------------------------------------------------------------------------
__global__ __launch_bounds__(256) void k1_gemm1(
    const float* __restrict__ vf, const int* __restrict__ coords,
    const float* __restrict__ pf, const float* __restrict__ W1,
    const float* __restrict__ b1, float* __restrict__ hT,
    float* __restrict__ partials, int N, int NBLK)
{
    __shared__ float lw1[DIN * HCH];   // 16 KB: W1[d][k]
    __shared__ float sred[8][2][HCH];  // per-wave, per-M-half partial sums
    __shared__ float qred[8][2][HCH];  // per-wave, per-M-half partial sumsq

    const int tid  = threadIdx.x;
    const int b    = blockIdx.y;
    const int blk  = blockIdx.x;

    // cooperative W1 -> LDS
    for (int i = tid; i < DIN * HCH; i += 256) lw1[i] = W1[i];
    __syncthreads();

    const int wave = tid >> 5;
    const int lane = tid & 31;
    const int m    = lane & 15;   // M row (A) / N column (B,D)
    const int kh   = lane >> 4;   // K/M half selector per WMMA f32 layout
    const int pt0  = (blk * 8 + wave) * 16;

    float s0 = 0.f, q0 = 0.f, s1 = 0.f, q1 = 0.f;

    if (pt0 < N) {  // wave-uniform: EXEC stays all-ones inside (WMMA requirement)
        int p = pt0 + m; if (p >= N) p = N - 1;
        const int* cp = coords + ((size_t)b * N + p) * 3;
        const int xi = cp[0], yi = cp[1], zi = cp[2];
        const size_t flat = ((size_t)zi * YD + yi) * XD + xi;
        const float* vbase = vf + (size_t)b * CCH * ZYX + flat;
        const float* pfb   = pf + ((size_t)b * N + p) * PCH;

        v8f acc0 = {}; // channels 0..15
        v8f acc1 = {}; // channels 16..31

        // K-steps 0..23: voxel channels (d < 96), strided gathers (L2-resident)
        #pragma unroll 4
        for (int kk = 0; kk < 24; ++kk) {
            const int d0 = kk * 4 + kh * 2;
            v2f a;
            a.x = vbase[(size_t)d0 * ZYX];
            a.y = vbase[(size_t)(d0 + 1) * ZYX];
            v2f bA, bB;
            bA.x = lw1[d0 * HCH + m];
            bA.y = lw1[(d0 + 1) * HCH + m];
            bB.x = lw1[d0 * HCH + 16 + m];
            bB.y = lw1[(d0 + 1) * HCH + 16 + m];
            acc0 = __builtin_amdgcn_wmma_f32_16x16x4_f32(false, a, false, bA,
                                                         (short)0, acc0, false, false);
            acc1 = __builtin_amdgcn_wmma_f32_16x16x4_f32(false, a, false, bB,
                                                         (short)0, acc1, false, false);
        }
        // K-steps 24..31: point-feat channels (contiguous float2)
        #pragma unroll
        for (int kk = 24; kk < 32; ++kk) {
            const int d0 = kk * 4 + kh * 2;
            v2f a = *(const v2f*)(pfb + (d0 - CCH));
            v2f bA, bB;
            bA.x = lw1[d0 * HCH + m];
            bA.y = lw1[(d0 + 1) * HCH + m];
            bB.x = lw1[d0 * HCH + 16 + m];
            bB.y = lw1[(d0 + 1) * HCH + 16 + m];
            acc0 = __builtin_amdgcn_wmma_f32_16x16x4_f32(false, a, false, bA,
                                                         (short)0, acc0, false, false);
            acc1 = __builtin_amdgcn_wmma_f32_16x16x4_f32(false, a, false, bB,
                                                         (short)0, acc1, false, false);
        }

        // bias, stats, store (D layout: vgpr r -> M = r + 8*kh, N = lane&15)
        const float biasA = b1[m];
        const float biasB = b1[16 + m];
        const int   mb    = pt0 + 8 * kh;
        float* outA = hT + ((size_t)b * HCH + m)      * (size_t)N + mb;
        float* outB = hT + ((size_t)b * HCH + 16 + m) * (size_t)N + mb;

        float hA[8], hB[8];
        #pragma unroll
        for (int r = 0; r < 8; ++r) {
            hA[r] = acc0[r] + biasA;
            hB[r] = acc1[r] + biasB;
            const float msk = (mb + r < N) ? 1.f : 0.f;
            s0 += hA[r] * msk; q0 += hA[r] * hA[r] * msk;
            s1 += hB[r] * msk; q1 += hB[r] * hB[r] * msk;
        }
        if (mb + 8 <= N) {
            v4f a0 = {hA[0], hA[1], hA[2], hA[3]};
            v4f a1 = {hA[4], hA[5], hA[6], hA[7]};
            v4f c0 = {hB[0], hB[1], hB[2], hB[3]};
            v4f c1 = {hB[4], hB[5], hB[6], hB[7]};
            *(v4f*)(outA) = a0; *(v4f*)(outA + 4) = a1;
            *(v4f*)(outB) = c0; *(v4f*)(outB + 4) = c1;
        } else {
            for (int r = 0; r < 8; ++r)
                if (mb + r < N) { outA[r] = hA[r]; outB[r] = hB[r]; }
        }
    }

    // deterministic block-level stats tree
    sred[wave][kh][m]      = s0;  sred[wave][kh][16 + m] = s1;
    qred[wave][kh][m]      = q0;  qred[wave][kh][16 + m] = q1;
    __syncthreads();
    if (tid < HCH) {
        float s = 0.f, q = 0.f;
        #pragma unroll
        for (int w = 0; w < 8; ++w)
            #pragma unroll
            for (int h = 0; h < 2; ++h) { s += sred[w][h][tid]; q += qred[w][h][tid]; }
        float* pp = partials + ((size_t)b * NBLK + blk) * 64;
        pp[tid]      = s;
        pp[32 + tid] = q;
    }
}

// ---------------------------------------------------------------------------
// Kernel 2: fold per-block partials (fixed order -> deterministic),
// produce per-(b,k) BN scale/shift: a = gamma*rsqrt(var+eps), c = beta - mean*a
// ---------------------------------------------------------------------------
__global__ void k2_stats(const float* __restrict__ partials,
                         const float* __restrict__ gamma,
                         const float* __restrict__ beta,
                         float* __restrict__ stats, int N, int NBLK)
{
    const int t = threadIdx.x;
    if (t >= BVAL * HCH) return;
    const int b = t >> 5, k = t & 31;
    float s = 0.f, q = 0.f;
    const float* pp = partials + (size_t)b * NBLK * 64;
    for (int i = 0; i < NBLK; ++i) { s += pp[i * 64 + k]; q += pp[i * 64 + 32 + k]; }
    const float invN = 1.f / (float)N;
    const float mean = s * invN;
    const float var  = fmaxf(q * invN - mean * mean, 0.f);
    const float a    = gamma[k] * rsqrtf(var + BN_EPS);
    stats[b * 64 + k]      = a;
    stats[b * 64 + 32 + k] = beta[k] - mean * a;
}

// ---------------------------------------------------------------------------
// Kernel 3: flow = relu(a*h + c) @ W2 + b2 (coalesced channel-major h reads;
// N-dim of 3 is too thin for WMMA, VALU fma is bandwidth-optimal here)
// ---------------------------------------------------------------------------
__global__ __launch_bounds__(256) void k3_flow(
    const float* __restrict__ hT, const float* __restrict__ stats,
    const float* __restrict__ W2, const float* __restrict__ b2,
    float* __restrict__ out, int N)
{
    __shared__ float la[HCH], lc[HCH], lw2[HCH * 3];
    const int tid = threadIdx.x;
    const int b   = blockIdx.y;
    if (tid < HCH) { la[tid] = stats[b * 64 + tid]; lc[tid] = stats[b * 64 + 32 + tid]; }
    if (tid < HCH * 3) lw2[tid] = W2[tid];
    __syncthreads();

    const int p = blockIdx.x * 256 + tid;
    if (p >= N) return;

    float f0 = b2[0], f1 = b2[1], f2 = b2[2];
    const float* hp = hT + (size_t)b * HCH * N + p;
    #pragma unroll
    for (int k = 0; k < HCH; ++k) {
        const float y = fmaxf(fmaf(la[k], hp[(size_t)k * N], lc[k]), 0.f);
        f0 = fmaf(y, lw2[k * 3 + 0], f0);
        f1 = fmaf(y, lw2[k * 3 + 1], f1);
        f2 = fmaf(y, lw2[k * 3 + 2], f2);
    }
    float* o = out + ((size_t)b * N + p) * 3;
    o[0] = f0; o[1] = f1; o[2] = f2;
}

// ---------------------------------------------------------------------------
// launcher
// ---------------------------------------------------------------------------
extern "C" void kernel_launch(void* const* d_in, const int* in_sizes, int n_in,
                              void* d_out, int out_size, void* d_ws, size_t ws_size,
                              hipStream_t stream) {
    const float* vf     = (const float*)d_in[0];  // [B,96,16,256,256]
    const int*   coords = (const int*)  d_in[1];  // [B,N,3]
    const float* pf     = (const float*)d_in[2];  // [B,N,32]
    const float* W1     = (const float*)d_in[3];  // [128,32]
    const float* b1     = (const float*)d_in[4];  // [32]
    const float* gamma  = (const float*)d_in[5];  // [32]
    const float* beta   = (const float*)d_in[6];  // [32]
    const float* W2     = (const float*)d_in[7];  // [32,3]
    const float* b2     = (const float*)d_in[8];  // [3]
    float* out = (float*)d_out;                   // [B,N,3]

    const int B    = BVAL;
    const int N    = in_sizes[1] / (B * 3);       // 100000
    const int NBLK = (N + 127) / 128;             // 16-pt tiles x 8 waves

    // workspace layout (~26 MB): hT | partials | stats
    float* hT       = (float*)d_ws;                       // B*32*N
    float* partials = hT + (size_t)B * HCH * N;           // B*NBLK*64
    float* stats    = partials + (size_t)B * NBLK * 64;   // B*64

    k1_gemm1<<<dim3(NBLK, B), 256, 0, stream>>>(vf, coords, pf, W1, b1,
                                                hT, partials, N, NBLK);
    k2_stats<<<1, 64, 0, stream>>>(partials, gamma, beta, stats, N, NBLK);
    k3_flow<<<dim3((N + 255) / 256, B), 256, 0, stream>>>(hT, stats, W2, b2, out, N);
}